// DepthwiseTensorProductModule_62148176773673
// MI455X (gfx1250) — compile-verified
//
#include <hip/hip_runtime.h>
#include <math.h>

#define NTHREADS 256
#define EPB 8

typedef int v4i __attribute__((ext_vector_type(4)));
typedef __attribute__((address_space(1))) v4i gv4i;
typedef __attribute__((address_space(3))) v4i lv4i;
typedef __attribute__((address_space(1))) int gi32;
typedef __attribute__((address_space(3))) int li32;

// ---------------- gfx1250 async global->LDS copy helpers ----------------
__device__ __forceinline__ void async_copy_b128(const void* g, void* l) {
#if __has_builtin(__builtin_amdgcn_global_load_async_to_lds_b128)
  __builtin_amdgcn_global_load_async_to_lds_b128((gv4i*)g, (lv4i*)l, 0, 0);
#else
  unsigned lo = (unsigned)(__UINTPTR_TYPE__)(li32*)l;
  unsigned long long ga = (unsigned long long)(__UINTPTR_TYPE__)g;
  asm volatile("global_load_async_to_lds_b128 %0, %1, off" :: "v"(lo), "v"(ga) : "memory");
#endif
}

__device__ __forceinline__ void async_copy_b32(const void* g, void* l) {
#if __has_builtin(__builtin_amdgcn_global_load_async_to_lds_b32)
  __builtin_amdgcn_global_load_async_to_lds_b32((gi32*)g, (li32*)l, 0, 0);
#else
  unsigned lo = (unsigned)(__UINTPTR_TYPE__)(li32*)l;
  unsigned long long ga = (unsigned long long)(__UINTPTR_TYPE__)g;
  asm volatile("global_load_async_to_lds_b32 %0, %1, off" :: "v"(lo), "v"(ga) : "memory");
#endif
}

__device__ __forceinline__ void wait_async0() {
#if __has_builtin(__builtin_amdgcn_s_wait_asynccnt)
  __builtin_amdgcn_s_wait_asynccnt(0);
#else
  asm volatile("s_wait_asynccnt 0" ::: "memory");
#endif
}

// non-temporal f32 store (output is streamed exactly once; keep it out of L2)
__device__ __forceinline__ void nt_store(float* p, float v) {
  __builtin_nontemporal_store(v, p);
}

// ---------------- CG table init (device transcription of reference math) ----------------
// NOTE: the compiler constant-folds all of this at compile time (verified in disasm:
// the kernel stores immediate float constants), so this is a one-wave table write.
__device__ double fct(int n) { double r = 1.0; for (int i = 2; i <= n; ++i) r *= (double)i; return r; }

__device__ double su2cg(int j1, int j2, int j3, int m1, int m2, int m3) {
  if (m3 != m1 + m2) return 0.0;
  int vmin = -j1 + j2 + m3; int t = -j1 + m1; if (t > vmin) vmin = t; if (vmin < 0) vmin = 0;
  int vmax = j2 + j3 + m1; t = j3 - j1 + j2; if (t < vmax) vmax = t; t = j3 + m3; if (t < vmax) vmax = t;
  double c = sqrt((double)(2 * j3 + 1) * fct(j3 + j1 - j2) * fct(j3 - j1 + j2) * fct(j1 + j2 - j3)
                  * fct(j3 + m3) * fct(j3 - m3)
                  / (fct(j1 + j2 + j3 + 1) * fct(j1 - m1) * fct(j1 + m1) * fct(j2 - m2) * fct(j2 + m2)));
  double s = 0.0;
  for (int v = vmin; v <= vmax; ++v) {
    double term = fct(j2 + j3 + m1 - v) * fct(j1 - m1 + v)
                / (fct(v) * fct(j3 - j1 + j2 - v) * fct(j3 + m3 - v) * fct(v + j1 - j2 - m3));
    s += ((v + j2 + m2) & 1) ? -term : term;
  }
  return c * s;
}

__device__ void rtc(int l, double* qre, double* qim) {
  int D = 2 * l + 1;
  for (int i = 0; i < D * D; ++i) { qre[i] = 0.0; qim[i] = 0.0; }
  const double is2 = 0.70710678118654752440;
  for (int m = -l; m < 0; ++m) {
    qre[(l + m) * D + (l - m)] = is2;     // q[l+m, l+|m|] = 1/sqrt(2)
    qim[(l + m) * D + (l + m)] = -is2;    // q[l+m, l-|m|] = -i/sqrt(2)
  }
  qre[l * D + l] = 1.0;
  for (int m = 1; m <= l; ++m) {
    double s = (m & 1) ? -1.0 : 1.0;      // (-1)^m
    qre[(l + m) * D + (l + m)] = s * is2;
    qim[(l + m) * D + (l - m)] = s * is2; // i*(-1)^m/sqrt(2)
  }
  double pr, pi;                          // (-i)^l
  if (l == 0) { pr = 1.0; pi = 0.0; } else if (l == 1) { pr = 0.0; pi = -1.0; } else { pr = -1.0; pi = 0.0; }
  for (int i = 0; i < D * D; ++i) {
    double re = qre[i] * pr - qim[i] * pi;
    double im = qre[i] * pi + qim[i] * pr;
    qre[i] = re; qim[i] = im;
  }
}

__device__ void so3cg_write(int l1, int l2, int l3, double alpha, float* out) {
  const int D1 = 2 * l1 + 1, D2 = 2 * l2 + 1, D3 = 2 * l3 + 1;
  double Cc[125];
  for (int i = 0; i < D1 * D2 * D3; ++i) Cc[i] = 0.0;
  for (int m1 = -l1; m1 <= l1; ++m1)
    for (int m2 = -l2; m2 <= l2; ++m2) {
      int m3 = m1 + m2;
      if (m3 >= -l3 && m3 <= l3)
        Cc[((l1 + m1) * D2 + (l2 + m2)) * D3 + (l3 + m3)] = su2cg(l1, l2, l3, m1, m2, m3);
    }
  double q1re[25], q1im[25], q2re[25], q2im[25], q3re[25], q3im[25];
  rtc(l1, q1re, q1im); rtc(l2, q2re, q2im); rtc(l3, q3re, q3im);
  double Cr[125]; double nrm = 0.0;
  for (int j = 0; j < D1; ++j)
    for (int lb = 0; lb < D2; ++lb)
      for (int mm = 0; mm < D3; ++mm) {
        double acc = 0.0;
        for (int i = 0; i < D1; ++i)
          for (int k = 0; k < D2; ++k)
            for (int n = 0; n < D3; ++n) {
              double cc = Cc[(i * D2 + k) * D3 + n];
              if (cc == 0.0) continue;
              double t1re = q1re[i * D1 + j],  t1im = q1im[i * D1 + j];
              double t2re = q2re[k * D2 + lb], t2im = q2im[k * D2 + lb];
              double t3re = q3re[n * D3 + mm], t3im = -q3im[n * D3 + mm]; // conj
              double pre = t1re * t2re - t1im * t2im;
              double pim = t1re * t2im + t1im * t2re;
              double rre = pre * t3re - pim * t3im;
              acc += rre * cc;
            }
        int idx = (j * D2 + lb) * D3 + mm;
        Cr[idx] = acc; nrm += acc * acc;
      }
  nrm = sqrt(nrm);
  double sc = alpha / nrm;
  for (int i = 0; i < D1 * D2 * D3; ++i) out[i] = (float)(Cr[i] * sc);
}

// ws layout (floats): C10@0(9) C11@9(27) C12@36(45) C20@81(25) C21@106(75) C22@181(125) = 306
__global__ void init_cg_kernel(float* __restrict__ cg) {
  if (threadIdx.x == 0 && blockIdx.x == 0) {
    const double a2 = sqrt(5.0 / 3.0); // alpha for l=2 outputs; alpha=1 for l=0,1
    so3cg_write(1, 0, 1, 1.0, cg + 0);
    so3cg_write(1, 1, 1, 1.0, cg + 9);
    so3cg_write(1, 2, 1, 1.0, cg + 36);
    so3cg_write(2, 0, 2, a2,  cg + 81);
    so3cg_write(2, 1, 2, a2,  cg + 106);
    so3cg_write(2, 2, 2, a2,  cg + 181);
  }
}

// ---------------- main kernel: one edge per wave, 8 edges per block ----------------
// x1 row: [l0:128 | l1:64x3 | l2:32x5] = 480 ; x2 row: [1|3|5] = 9 ; w row: [128|64|64|64|32|32|32] = 416
// out row: [128 | 64x3 | 32x5] = 480
__global__ __launch_bounds__(NTHREADS) void dtp_kernel(
    const float* __restrict__ x1g, const float* __restrict__ x2g,
    const float* __restrict__ wg, const float* __restrict__ cg,
    float* __restrict__ outg, int E) {
  __shared__ __align__(16) float sX1[EPB * 480];
  __shared__ __align__(16) float sW[EPB * 416];
  __shared__ __align__(16) float sX2[EPB * 9];
  // per-edge M buffer: [M10(9) M11(9) M12(9) M20(25) M21(25) M22(25)] = 102, stride 104
  __shared__ __align__(16) float sM[EPB * 104];

  const int tid = threadIdx.x;
  const int e0 = blockIdx.x * EPB;
  int nE = E - e0; if (nE > EPB) nE = EPB;
  if (nE <= 0) return;

  // ---- stage x1 / w / x2 tiles via gfx1250 async copies (contiguous rows) ----
  {
    const char* g1 = (const char*)(x1g + (size_t)e0 * 480);
    const int n1 = nE * 120; // b128 chunks (480*4/16 per edge)
    for (int i = tid; i < n1; i += NTHREADS)
      async_copy_b128(g1 + (size_t)i * 16, (char*)sX1 + (size_t)i * 16);

    const char* g2 = (const char*)(wg + (size_t)e0 * 416);
    const int n2 = nE * 104; // 416*4/16
    for (int i = tid; i < n2; i += NTHREADS)
      async_copy_b128(g2 + (size_t)i * 16, (char*)sW + (size_t)i * 16);

    const char* g3 = (const char*)(x2g + (size_t)e0 * 9);
    const int n3 = nE * 9;   // b32 chunks (only 4B-aligned)
    for (int i = tid; i < n3; i += NTHREADS)
      async_copy_b32(g3 + (size_t)i * 4, (char*)sX2 + (size_t)i * 4);
  }
  wait_async0();
  __syncthreads();

  // ---- build per-edge M_p[a,c] = sum_b C_p[a,b,c] * x2_p[b] cooperatively ----
  {
    const int total = nE * 102;
    for (int idx = tid; idx < total; idx += NTHREADS) {
      int e = idx / 102;
      int r = idx - e * 102;
      int base, d, bdim, x2off, coff;
      if (r < 27) {
        d = 3;
        if (r < 9)       { base = 0;  bdim = 1; x2off = 0; coff = 0;  }
        else if (r < 18) { base = 9;  bdim = 3; x2off = 1; coff = 9;  }
        else             { base = 18; bdim = 5; x2off = 4; coff = 36; }
      } else {
        d = 5;
        if (r < 52)      { base = 27; bdim = 1; x2off = 0; coff = 81;  }
        else if (r < 77) { base = 52; bdim = 3; x2off = 1; coff = 106; }
        else             { base = 77; bdim = 5; x2off = 4; coff = 181; }
      }
      int loc = r - base;
      int a = loc / d;
      int c = loc - a * d;
      float acc = 0.0f;
      for (int b = 0; b < bdim; ++b)
        acc += cg[coff + (a * bdim + b) * d + c] * sX2[e * 9 + x2off + b];
      sM[e * 104 + r] = acc;
    }
  }
  __syncthreads();

  // ---- compute: wave w handles edge e0+w ----
  const int w = tid >> 5;
  const int lane = tid & 31;
  if (w < nE) {
    const float* X1 = &sX1[w * 480];
    const float* Wt = &sW[w * 416];
    const float* M  = &sM[w * 104];
    float* orow = outg + (size_t)(e0 + w) * 480;
    const float x2s0 = sX2[w * 9];

    // out0 (l=0, 128 channels): C000=1, alpha=1
#pragma unroll
    for (int r = 0; r < 4; ++r) {
      int u = lane + 32 * r;
      nt_store(&orow[u], X1[u] * Wt[u] * x2s0);
    }

    // out1 (l=1, 64 channels x 3)
    {
      float m10d[3], m11[9], m12[9];
#pragma unroll
      for (int i = 0; i < 3; ++i) m10d[i] = M[i * 3 + i];  // l2=0 path is diagonal
#pragma unroll
      for (int i = 0; i < 9; ++i) { m11[i] = M[9 + i]; m12[i] = M[18 + i]; }
#pragma unroll
      for (int r = 0; r < 2; ++r) {
        int u = lane + 32 * r;
        float a0 = X1[128 + 3 * u + 0];
        float a1 = X1[128 + 3 * u + 1];
        float a2 = X1[128 + 3 * u + 2];
        float w0 = Wt[128 + u], w1 = Wt[192 + u], w2 = Wt[256 + u];
#pragma unroll
        for (int c = 0; c < 3; ++c) {
          float mu0 = w1 * m11[0 + c] + w2 * m12[0 + c];
          float mu1 = w1 * m11[3 + c] + w2 * m12[3 + c];
          float mu2 = w1 * m11[6 + c] + w2 * m12[6 + c];
          float acc = a0 * mu0 + a1 * mu1 + a2 * mu2;
          float xc = (c == 0) ? a0 : ((c == 1) ? a1 : a2);
          acc += w0 * m10d[c] * xc;
          nt_store(&orow[128 + 3 * u + c], acc);
        }
      }
    }

    // out2 (l=2, 32 channels x 5); alpha=sqrt(5/3) folded into C tables
    {
      float m20d[5], m21r[25], m22r[25];
#pragma unroll
      for (int i = 0; i < 5; ++i) m20d[i] = M[27 + i * 5 + i];
#pragma unroll
      for (int i = 0; i < 25; ++i) { m21r[i] = M[52 + i]; m22r[i] = M[77 + i]; }
      int u = lane;
      float xa[5];
#pragma unroll
      for (int j = 0; j < 5; ++j) xa[j] = X1[320 + 5 * u + j];
      float w0 = Wt[320 + u], w1 = Wt[352 + u], w2 = Wt[384 + u];
#pragma unroll
      for (int c = 0; c < 5; ++c) {
        float acc = w0 * m20d[c] * xa[c];
#pragma unroll
        for (int a = 0; a < 5; ++a)
          acc += xa[a] * (w1 * m21r[a * 5 + c] + w2 * m22r[a * 5 + c]);
        nt_store(&orow[320 + 5 * u + c], acc);
      }
    }
  }
}

extern "C" void kernel_launch(void* const* d_in, const int* in_sizes, int n_in,
                              void* d_out, int out_size, void* d_ws, size_t ws_size,
                              hipStream_t stream) {
  const float* x1 = (const float*)d_in[0];
  const float* x2 = (const float*)d_in[1];
  const float* wt = (const float*)d_in[2];
  float* out = (float*)d_out;
  float* cg = (float*)d_ws;
  const int E = in_sizes[0] / 480;

  init_cg_kernel<<<1, 32, 0, stream>>>(cg);
  const int blocks = (E + EPB - 1) / EPB;
  dtp_kernel<<<blocks, NTHREADS, 0, stream>>>(x1, x2, wt, cg, out, E);
}